// NeuSRenderer_32753420599831
// MI455X (gfx1250) — compile-verified
//
#include <hip/hip_runtime.h>
#include <cstdint>
#include <cstddef>

// NeuS volume-rendering forward pass for MI455X (gfx1250).
// Memory-bound streaming scan: async global->LDS double-buffered pipeline
// (CDNA5 GLOBAL_LOAD_ASYNC_TO_LDS + ASYNCcnt), wave32-aware, coalesced b64.

namespace {

constexpr int kB = 2;
constexpr int kD = 128;
constexpr int kH = 256;
constexpr int kW = 256;
constexpr int kHW = kH * kW;                    // 65536
constexpr int kBlock = 256;                     // 8 wave32s
constexpr int kWavesPerBlock = kBlock / 32;     // 8
constexpr int kRaysPerWave = 64;                // 2 rays per lane
constexpr int kT = 4;                           // depth rows per tile
constexpr int kNT = kD / kT;                    // 32 tiles
// LDS layout (bytes): [wave][buf:2][stream:4][row:kT][64 rays * 4B]
constexpr int kRowB    = kRaysPerWave * 4;      // 256
constexpr int kStreamB = kT * kRowB;            // 1024
constexpr int kBufB    = 4 * kStreamB;          // 4096
constexpr int kWaveB   = 2 * kBufB;             // 8192
constexpr int kSmemB   = kWavesPerBlock * kWaveB; // 65536 (<= 320KB WGP LDS)

constexpr float kEpsAlpha = 1e-5f;
constexpr float kEpsTrans = 1e-7f;
constexpr float kLog2e    = 1.44269504088896340736f;

// CDNA5 async copy: per-lane 8B global -> LDS, tracked by ASYNCcnt.
__device__ __forceinline__ void asyncLoadB64(uint32_t ldsAddr, int voff,
                                             const float* sbase) {
  asm volatile("global_load_async_to_lds_b64 %0, %1, %2"
               :: "v"(ldsAddr), "v"(voff), "s"(sbase)
               : "memory");
}

template <int N>
__device__ __forceinline__ void waitAsync() {
  asm volatile("s_wait_asynccnt %0" :: "n"(N) : "memory");
}

// Issue one tile: kT depth rows x 4 streams = 16 async b64 ops per wave.
__device__ __forceinline__ void issueTile(uint32_t ldsWaveLane, int buf, int tile,
                                          int lane,
                                          const float* s0, const float* s1,
                                          const float* s2, const float* s3) {
  const uint32_t base = ldsWaveLane + (uint32_t)buf * kBufB;
#pragma unroll
  for (int j = 0; j < kT; ++j) {
    const int d = tile * kT + j;
    const int voff = d * (kHW * 4) + lane * 8;   // byte offset, fits i32
    asyncLoadB64(base + 0 * kStreamB + j * kRowB, voff, s0);
    asyncLoadB64(base + 1 * kStreamB + j * kRowB, voff, s1);
    asyncLoadB64(base + 2 * kStreamB + j * kRowB, voff, s2);
    asyncLoadB64(base + 3 * kStreamB + j * kRowB, voff, s3);
  }
}

__device__ __forceinline__ float sigmoidFast(float x, float k) {
  // 1 / (1 + exp(-x*invS)) with exp via v_exp_f32 (exp2), rcp via v_rcp_f32.
  return __builtin_amdgcn_rcpf(1.0f + __builtin_amdgcn_exp2f(-x * k));
}

} // namespace

__global__ __launch_bounds__(kBlock)
void neus_render_kernel(const float* __restrict__ sdf,
                        const float* __restrict__ color,
                        const float* __restrict__ variance,
                        float* __restrict__ out) {
  __shared__ float smem[kSmemB / 4];

  const int lane = threadIdx.x & 31;
  const int wv   = threadIdx.x >> 5;
  const int gw   = blockIdx.x * kWavesPerBlock + wv;   // global wave id
  const int r0   = gw * kRaysPerWave;                  // wave's first ray
  const int b    = r0 / kHW;                           // 64 | kHW -> uniform
  const int hw0  = r0 % kHW;

  const float* sSdf = sdf + (size_t)b * kD * kHW + hw0;
  const float* sR   = color + ((size_t)b * 3 + 0) * ((size_t)kD * kHW) + hw0;
  const float* sG   = color + ((size_t)b * 3 + 1) * ((size_t)kD * kHW) + hw0;
  const float* sB   = color + ((size_t)b * 3 + 2) * ((size_t)kD * kHW) + hw0;

  // Low 32 bits of a generic LDS pointer == LDS byte address on gfx1250.
  const uint32_t smemBase    = (uint32_t)(uintptr_t)(&smem[0]);
  const uint32_t ldsWaveLane = smemBase + (uint32_t)wv * kWaveB + (uint32_t)lane * 8;

  const float invS = __builtin_amdgcn_exp2f(variance[0] * (10.0f * kLog2e));
  const float kExp = invS * kLog2e;   // exp(-s*invS) == exp2(-s*kExp)

  // Prime the pipeline with tile 0 in buffer 0.
  issueTile(ldsWaveLane, 0, 0, lane, sSdf, sR, sG, sB);

  // Per-ray scan state (2 rays per thread).
  float trans0 = 1.0f, trans1 = 1.0f;
  float pc0 = 0.0f, pc1 = 0.0f;                 // previous sigmoid CDF
  float pr0 = 0, pr1 = 0, pg0 = 0, pg1 = 0, pb0 = 0, pb1 = 0;  // previous color
  float ar0 = 0, ar1 = 0, ag0 = 0, ag1 = 0, ab0 = 0, ab1 = 0;  // color accum
  float ws0 = 0, ws1 = 0;                        // weight sum

  const float* waveSmem = smem + (wv * kWaveB) / 4;

  for (int t = 0; t < kNT; ++t) {
    if (t + 1 < kNT) {
      issueTile(ldsWaveLane, (t + 1) & 1, t + 1, lane, sSdf, sR, sG, sB);
      waitAsync<16>();   // 16 newest pending -> tile t fully resident (in-order)
    } else {
      waitAsync<0>();
    }

    const float* buf = waveSmem + ((t & 1) * kBufB) / 4;
#pragma unroll
    for (int j = 0; j < kT; ++j) {
      const int d = t * kT + j;
      const int rowOff = (j * kRowB) / 4 + lane * 2;
      const float2 sv = *(const float2*)(buf + 0 * (kStreamB / 4) + rowOff);
      const float2 rv = *(const float2*)(buf + 1 * (kStreamB / 4) + rowOff);
      const float2 gv = *(const float2*)(buf + 2 * (kStreamB / 4) + rowOff);
      const float2 bv = *(const float2*)(buf + 3 * (kStreamB / 4) + rowOff);

      const float c0 = sigmoidFast(sv.x, kExp);
      const float c1 = sigmoidFast(sv.y, kExp);

      if (d != 0) {   // step i = d-1 of the reference scan
        float a0 = (pc0 - c0 + kEpsAlpha) * __builtin_amdgcn_rcpf(pc0 + kEpsAlpha);
        a0 = fminf(fmaxf(a0, 0.0f), 1.0f);
        const float w0 = a0 * trans0;
        ar0 = fmaf(pr0, w0, ar0); ag0 = fmaf(pg0, w0, ag0); ab0 = fmaf(pb0, w0, ab0);
        ws0 += w0;
        trans0 *= (1.0f - a0 + kEpsTrans);

        float a1 = (pc1 - c1 + kEpsAlpha) * __builtin_amdgcn_rcpf(pc1 + kEpsAlpha);
        a1 = fminf(fmaxf(a1, 0.0f), 1.0f);
        const float w1 = a1 * trans1;
        ar1 = fmaf(pr1, w1, ar1); ag1 = fmaf(pg1, w1, ag1); ab1 = fmaf(pb1, w1, ab1);
        ws1 += w1;
        trans1 *= (1.0f - a1 + kEpsTrans);
      }
      pc0 = c0; pc1 = c1;
      pr0 = rv.x; pr1 = rv.y;
      pg0 = gv.x; pg1 = gv.y;
      pb0 = bv.x; pb1 = bv.y;
    }
  }

  // Outputs: color (B,3,H,W) then weights_sum (B,1,H,W), concatenated flat.
  const int oc = b * 3 * kHW + hw0 + lane * 2;
  *(float2*)(out + oc + 0 * kHW) = make_float2(ar0, ar1);
  *(float2*)(out + oc + 1 * kHW) = make_float2(ag0, ag1);
  *(float2*)(out + oc + 2 * kHW) = make_float2(ab0, ab1);
  *(float2*)(out + kB * 3 * kHW + b * kHW + hw0 + lane * 2) = make_float2(ws0, ws1);
}

extern "C" void kernel_launch(void* const* d_in, const int* in_sizes, int n_in,
                              void* d_out, int out_size, void* d_ws, size_t ws_size,
                              hipStream_t stream) {
  (void)in_sizes; (void)n_in; (void)out_size; (void)d_ws; (void)ws_size;
  const float* sdf = (const float*)d_in[0];
  const float* col = (const float*)d_in[1];
  const float* var = (const float*)d_in[2];
  float* out = (float*)d_out;

  const int totalRays = kB * kHW;                          // 131072
  const int blocks = totalRays / (kWavesPerBlock * kRaysPerWave); // 256
  neus_render_kernel<<<dim3(blocks), dim3(kBlock), 0, stream>>>(sdf, col, var, out);
}